// BiLSTM_45612552684163
// MI455X (gfx1250) — compile-verified
//
#include <hip/hip_runtime.h>
#include <math.h>

// ---------------------------------------------------------------------------
// Model dims (fixed by the reference)
// ---------------------------------------------------------------------------
#define T_SEQ   512      // N_R == N_L == 512
#define HID     250      // LSTM hidden per direction
#define G4H     1000     // 4*HID
#define GX_LD   1024     // padded leading dim for gate buffers
#define HCAT_LD 512      // padded leading dim for [T, 2H] concat buffers (500 valid)
#define WHHT_LD 1024     // padded leading dim for transposed Whh  [250, 1024]
#define KC_LDS  64       // rows of WhhT cached in LDS (64*1024*4 = 256 KB of the 320KB WGP LDS)

typedef float v2f __attribute__((ext_vector_type(2)));
typedef float v8f __attribute__((ext_vector_type(8)));

__device__ __forceinline__ float fsigmoid(float x) {
    return 1.0f / (1.0f + __expf(-x));
}
__device__ __forceinline__ float ftanh_fast(float x) {
    // 1 - 2/(e^{2x}+1): stable at +/- inf (no inf/inf)
    float e = __expf(2.0f * x);
    return 1.0f - 2.0f / (e + 1.0f);
}

// ---------------------------------------------------------------------------
// Pre-transpose the 4 recurrent weight matrices Whh[1000,250] -> WhhT[250,1024]
// so the per-step matvec streams fully coalesced (thread j reads column j).
// ---------------------------------------------------------------------------
__global__ __launch_bounds__(256) void transpose_whh(
    const float* __restrict__ w0, const float* __restrict__ w1,
    const float* __restrict__ w2, const float* __restrict__ w3,
    float* __restrict__ outBase)
{
    int idx = blockIdx.x * 256 + threadIdx.x;
    const int per = G4H * HID;            // 250000
    if (idx >= 4 * per) return;
    int m   = idx / per;
    int rem = idx - m * per;
    int j   = rem / HID;                  // 0..999  (Whh row)
    int k   = rem - j * HID;              // 0..249  (Whh col)
    const float* src = (m == 0) ? w0 : (m == 1) ? w1 : (m == 2) ? w2 : w3;
    outBase[(size_t)m * (HID * WHHT_LD) + (size_t)k * WHHT_LD + j] = src[rem];
}

// ---------------------------------------------------------------------------
// fp32 WMMA GEMM:  C[M=512, N] = A[512, K] @ W[N, K]^T  (+bias1+bias2) (relu?)
// One wave32 computes a 16(M) x 64(N) strip = 4 accumulators fed by
// V_WMMA_F32_16X16X4_F32; the A fragment is loaded once and reused by all 4
// WMMAs per k-step (5 loads : 4 WMMAs), and the 4 independent WMMAs hide the
// accumulate latency. Block = 256 threads = 8 waves stacked along M;
// grid = (ceil(N/64), M/128).
// Fragment layouts (ISA 7.12.2):
//   A 16x4 f32 : lanes 0-15 -> rows m0+l (K=k,k+1); lanes 16-31 -> K=k+2,k+3
//   B 4x16 f32 : mirrored; column n0+l per half-wave
//   C  16x16   : VGPR p -> row m0+p (half 0) / m0+8+p (half 1), col n0+(lane&15)
// ---------------------------------------------------------------------------
__global__ __launch_bounds__(256) void gemm_wmma_f32(
    const float* __restrict__ A, int lda,
    const float* __restrict__ W, int ldw,
    const float* __restrict__ bias1,   // nullable
    const float* __restrict__ bias2,   // nullable
    float* __restrict__ C, int ldc,
    int N, int K, int relu)
{
    int tid  = threadIdx.x;
    int wave = tid >> 5;
    int lane = tid & 31;
    int half = lane >> 4;       // 0 or 1
    int l16  = lane & 15;
    int m0   = blockIdx.y * 128 + wave * 16;
    int n0   = blockIdx.x * 64;
    int koff = 2 * half;

    const float* aRow = A + (size_t)(m0 + l16) * lda + koff;

    bool wvalid[4];
    const float* wRow[4];
#pragma unroll
    for (int t = 0; t < 4; ++t) {
        int wr = n0 + 16 * t + l16;
        wvalid[t] = (wr < N);
        wRow[t] = W + (size_t)(wvalid[t] ? wr : 0) * ldw + koff;
    }

    v8f acc[4];
#pragma unroll
    for (int t = 0; t < 4; ++t)
#pragma unroll
        for (int i = 0; i < 8; ++i) acc[t][i] = 0.0f;

    if (n0 + 64 <= N) {
        // Fast path: whole 64-wide strip in range -> no masking at all.
        for (int k = 0; k < K; k += 4) {
            float2 av = *(const float2*)(aRow + k);
            v2f a; a[0] = av.x; a[1] = av.y;
#pragma unroll
            for (int t = 0; t < 4; ++t) {
                float2 bv = *(const float2*)(wRow[t] + k);
                v2f b; b[0] = bv.x; b[1] = bv.y;
                acc[t] = __builtin_amdgcn_wmma_f32_16x16x4_f32(
                             false, a, false, b, (short)0, acc[t], false, false);
            }
        }
    } else {
        // Edge path: loads are unconditional from a clamped (valid) row,
        // zeroed by cndmask -- no exec-mask branching in the hot loop.
        for (int k = 0; k < K; k += 4) {
            float2 av = *(const float2*)(aRow + k);
            v2f a; a[0] = av.x; a[1] = av.y;
#pragma unroll
            for (int t = 0; t < 4; ++t) {
                float2 bv = *(const float2*)(wRow[t] + k);
                v2f b;
                b[0] = wvalid[t] ? bv.x : 0.0f;
                b[1] = wvalid[t] ? bv.y : 0.0f;
                acc[t] = __builtin_amdgcn_wmma_f32_16x16x4_f32(
                             false, a, false, b, (short)0, acc[t], false, false);
            }
        }
    }

#pragma unroll
    for (int t = 0; t < 4; ++t) {
        int col = n0 + 16 * t + l16;
        if (col < N) {
            float badd = 0.0f;
            if (bias1) badd += bias1[col];
            if (bias2) badd += bias2[col];
#pragma unroll
            for (int p = 0; p < 8; ++p) {
                int row = m0 + half * 8 + p;
                float v = acc[t][p] + badd;
                if (relu) v = fmaxf(v, 0.0f);
                C[(size_t)row * ldc + col] = v;
            }
        }
    }
}

// ---------------------------------------------------------------------------
// Sequential LSTM recurrence for one layer. 4 blocks: b = seq*2 + dir.
// Block = 1024 threads (32 wave32 = full WGP). Per step:
//   g_j = gx[t,j] + sum_k h[k]*WhhT[k,j]   (j < 1000, coalesced over j)
// First KC_LDS k-rows of WhhT (256 KB) live in the 320 KB WGP LDS; the rest
// streams from L2. Gate threads (j < 250) own c in registers.
// ---------------------------------------------------------------------------
__global__ __launch_bounds__(1024) void lstm_layer(
    const float* __restrict__ gx_all,    // 4 x [512, GX_LD]
    const float* __restrict__ whhT_base, // 2 x [250, WHHT_LD] (f, b) for this layer
    float* __restrict__ hcat_base,       // 2 x [512, HCAT_LD]
    int T)
{
    extern __shared__ float smem[];
    float* hS   = smem;                  // 256
    float* gbuf = smem + 256;            // 1024
    float* whhC = smem + 1280;           // KC_LDS * WHHT_LD

    int b = blockIdx.x;
    int s = b >> 1;
    int d = b & 1;
    const float* gx   = gx_all    + (size_t)b * T_SEQ * GX_LD;
    const float* whhT = whhT_base + (size_t)d * HID * WHHT_LD;
    float*       hout = hcat_base + (size_t)s * T_SEQ * HCAT_LD + d * HID;

    int tid = threadIdx.x;
    for (int idx = tid; idx < KC_LDS * WHHT_LD; idx += 1024) whhC[idx] = whhT[idx];
    if (tid < HID) hS[tid] = 0.0f;
    float cst = 0.0f;
    __syncthreads();

    for (int t = 0; t < T; ++t) {
        int tt = d ? (T - 1 - t) : t;
        if (tid < G4H) {
            if (t + 1 < T) {
                int tn = d ? (T - 2 - t) : (t + 1);
                __builtin_prefetch(gx + (size_t)tn * GX_LD + tid, 0, 0);
            }
            float g = gx[(size_t)tt * GX_LD + tid];
            const float* wc = whhC + tid;
            const float* wg = whhT + tid;
#pragma unroll 8
            for (int k = 0; k < KC_LDS; ++k)
                g = fmaf(hS[k], wc[(size_t)k * WHHT_LD], g);
#pragma unroll 2
            for (int k = KC_LDS; k < HID; ++k)
                g = fmaf(hS[k], wg[(size_t)k * WHHT_LD], g);
            gbuf[tid] = g;
        }
        __syncthreads();
        if (tid < HID) {
            float gi = gbuf[tid];
            float gf = gbuf[tid + HID];
            float gg = gbuf[tid + 2 * HID];
            float go = gbuf[tid + 3 * HID];
            float i = fsigmoid(gi);
            float f = fsigmoid(gf);
            float g = ftanh_fast(gg);
            float o = fsigmoid(go);
            cst = f * cst + i * g;
            float h = o * ftanh_fast(cst);
            hS[tid] = h;
            hout[(size_t)tt * HCAT_LD + tid] = h;
        }
        __syncthreads();
    }
}

// ---------------------------------------------------------------------------
// Fused pairwise head: for every (r,l):
//   s_j     = relu(pr[r,j] + plb[l,j])           (b3 folded into plb's GEMM)
//   logit_k = sum_j s_j * Wout[k,j] + bout[k]
//   out     = log_softmax(logit)
// 16x16 pair tile per 256-thread block. pl tile + Wout in LDS (conflict-free
// 520-float stride, float4 reads); pr rows via broadcast-coalesced L0 hits.
// ---------------------------------------------------------------------------
__global__ __launch_bounds__(256) void pairwise_head(
    const float* __restrict__ pr,    // [512,512]
    const float* __restrict__ plb,   // [512,512] (pl + b3)
    const float* __restrict__ Wout,  // [2,512]
    const float* __restrict__ bout,  // [2]
    float* __restrict__ out)         // [512*512, 2]
{
    __shared__ float plS[16 * 520];
    __shared__ float w0S[512];
    __shared__ float w1S[512];

    int tid = threadIdx.x;
    int l0  = blockIdx.x * 16;
    int r0  = blockIdx.y * 16;

    for (int idx = tid; idx < 16 * 512; idx += 256) {
        int row = idx >> 9;
        int col = idx & 511;
        plS[row * 520 + col] = plb[(size_t)(l0 + row) * 512 + col];
    }
    for (int idx = tid; idx < 512; idx += 256) {
        w0S[idx] = Wout[idx];
        w1S[idx] = Wout[512 + idx];
    }
    __syncthreads();

    int tx = tid & 15;   // l within tile
    int ty = tid >> 4;   // r within tile
    const float* prRow = pr  + (size_t)(r0 + ty) * 512;
    const float* plRow = plS + tx * 520;

    float acc0 = 0.0f, acc1 = 0.0f;
    for (int j = 0; j < 512; j += 4) {
        float4 pv = *(const float4*)(prRow + j);
        float4 qv = *(const float4*)(plRow + j);
        float4 w0 = *(const float4*)(w0S + j);
        float4 w1 = *(const float4*)(w1S + j);
        float s0 = fmaxf(pv.x + qv.x, 0.0f);
        float s1 = fmaxf(pv.y + qv.y, 0.0f);
        float s2 = fmaxf(pv.z + qv.z, 0.0f);
        float s3 = fmaxf(pv.w + qv.w, 0.0f);
        acc0 = fmaf(s0, w0.x, acc0); acc1 = fmaf(s0, w1.x, acc1);
        acc0 = fmaf(s1, w0.y, acc0); acc1 = fmaf(s1, w1.y, acc1);
        acc0 = fmaf(s2, w0.z, acc0); acc1 = fmaf(s2, w1.z, acc1);
        acc0 = fmaf(s3, w0.w, acc0); acc1 = fmaf(s3, w1.w, acc1);
    }
    float lg0 = acc0 + bout[0];
    float lg1 = acc1 + bout[1];
    float m   = fmaxf(lg0, lg1);
    float lse = m + __logf(__expf(lg0 - m) + __expf(lg1 - m));
    size_t o  = ((size_t)(r0 + ty) * 512 + (size_t)(l0 + tx)) * 2;
    out[o]     = lg0 - lse;
    out[o + 1] = lg1 - lse;
}

// ---------------------------------------------------------------------------
// Host launcher (graph-capture safe: stream launches + one attribute set only)
// ---------------------------------------------------------------------------
extern "C" void kernel_launch(void* const* d_in, const int* in_sizes, int n_in,
                              void* d_out, int out_size, void* d_ws, size_t ws_size,
                              hipStream_t stream)
{
    (void)in_sizes; (void)n_in; (void)out_size; (void)ws_size;

    const float* v_r = (const float*)d_in[0];
    const float* v_l = (const float*)d_in[1];
    const float* Wih[4], *Whh[4], *bih[4], *bhh[4];
    for (int g = 0; g < 4; ++g) {           // l0f, l0b, l1f, l1b
        Wih[g] = (const float*)d_in[2 + 4 * g];
        Whh[g] = (const float*)d_in[3 + 4 * g];
        bih[g] = (const float*)d_in[4 + 4 * g];
        bhh[g] = (const float*)d_in[5 + 4 * g];
    }
    const float* W1   = (const float*)d_in[18];
    const float* b1   = (const float*)d_in[19];
    const float* W2   = (const float*)d_in[20];
    const float* b2   = (const float*)d_in[21];
    const float* W3   = (const float*)d_in[22];
    const float* b3   = (const float*)d_in[23];
    const float* Wout = (const float*)d_in[24];
    const float* bout = (const float*)d_in[25];

    // Workspace layout (floats); total 6,266,880 f32 ~= 25 MB
    float* ws    = (float*)d_ws;
    float* whhT  = ws;                                   // 4 * 250*1024   = 1,024,000
    float* gx    = whhT + 4 * (HID * WHHT_LD);           // 4 * 512*1024   = 2,097,152
    float* h0    = gx   + 4 * (T_SEQ * GX_LD);           // 2 * 512*512
    float* h1    = h0   + 2 * (T_SEQ * HCAT_LD);         // 2 * 512*512
    float* mlpa  = h1   + 2 * (T_SEQ * HCAT_LD);         // 2 * 512*1024
    float* rl2   = mlpa + 2 * (T_SEQ * 1024);            // 2 * 512*512
    float* prpl  = rl2  + 2 * (T_SEQ * 512);             // 2 * 512*512

    const int lstmSmem = (1280 + KC_LDS * WHHT_LD) * (int)sizeof(float); // 267,264 B
    (void)hipFuncSetAttribute(reinterpret_cast<const void*>(lstm_layer),
                              hipFuncAttributeMaxDynamicSharedMemorySize, lstmSmem);

    // 0) transpose recurrent weights
    transpose_whh<<<(4 * G4H * HID + 255) / 256, 256, 0, stream>>>(
        Whh[0], Whh[1], Whh[2], Whh[3], whhT);

    auto gemm = [&](const float* A, int lda, const float* W, int ldw,
                    const float* bb1, const float* bb2,
                    float* C, int ldc, int N, int K, int relu) {
        dim3 g((N + 63) / 64, T_SEQ / 128);
        gemm_wmma_f32<<<g, 256, 0, stream>>>(A, lda, W, ldw, bb1, bb2, C, ldc, N, K, relu);
    };

    const float* vin[2] = { v_r, v_l };

    // 1) layer-0 gate projections: gx = x @ Wih^T + (bih+bhh)   [512,1000], K=20
    for (int s = 0; s < 2; ++s)
        for (int d = 0; d < 2; ++d)
            gemm(vin[s], 20, Wih[d], 20, bih[d], bhh[d],
                 gx + (size_t)(s * 2 + d) * (T_SEQ * GX_LD), GX_LD, G4H, 20, 0);

    // 2) layer-0 recurrence -> h0 (cols 0..249 fwd, 250..499 bwd)
    lstm_layer<<<4, 1024, lstmSmem, stream>>>(gx, whhT, h0, T_SEQ);

    // 3) layer-1 gate projections from h0 [512,500], K=500
    for (int s = 0; s < 2; ++s)
        for (int d = 0; d < 2; ++d)
            gemm(h0 + (size_t)s * (T_SEQ * HCAT_LD), HCAT_LD,
                 Wih[2 + d], 2 * HID, bih[2 + d], bhh[2 + d],
                 gx + (size_t)(s * 2 + d) * (T_SEQ * GX_LD), GX_LD, G4H, 2 * HID, 0);

    // 4) layer-1 recurrence -> h1
    lstm_layer<<<4, 1024, lstmSmem, stream>>>(gx, whhT + 2 * (HID * WHHT_LD), h1, T_SEQ);

    // 5) branch MLP (shared weights, both sequences)
    for (int s = 0; s < 2; ++s) {
        gemm(h1 + (size_t)s * (T_SEQ * HCAT_LD), HCAT_LD, W1, 2 * HID, b1, nullptr,
             mlpa + (size_t)s * (T_SEQ * 1024), 1024, 1024, 2 * HID, 1);
        gemm(mlpa + (size_t)s * (T_SEQ * 1024), 1024, W2, 1024, b2, nullptr,
             rl2 + (size_t)s * (T_SEQ * 512), 512, 512, 1024, 1);
    }

    // 6) pr = r2 @ W3[:, :512]^T ; plb = l2 @ W3[:, 512:]^T + b3 (b3 folded here)
    gemm(rl2,                 512, W3,       1024, nullptr, nullptr, prpl,                 512, 512, 512, 0);
    gemm(rl2 + (T_SEQ * 512), 512, W3 + 512, 1024, b3,      nullptr, prpl + (T_SEQ * 512), 512, 512, 512, 0);

    // 7) fused pairwise relu + Wout + log_softmax -> d_out [512*512, 2]
    dim3 pg(32, 32);
    pairwise_head<<<pg, 256, 0, stream>>>(prpl, prpl + (T_SEQ * 512), Wout, bout, (float*)d_out);
}